// QuantumRegressionModel_65481071400512
// MI455X (gfx1250) — compile-verified
//
#include <hip/hip_runtime.h>
#include <cstdint>
#include <cstddef>

// ---------------- problem constants ----------------
constexpr int NWIRES  = 14;
constexpr int NLAYERS = 3;
constexpr int PPL     = 3;
constexpr int BATCH   = 1024;
constexpr int DIM     = 1 << NWIRES;   // 16384 amplitudes per state
constexpr int NT      = 512;           // 16 wave32 per block

// ---------------- circuit descriptor (host-built, kernarg) ----------------
struct PassDesc {
  uint32_t m[5];     // pair masks  = columns of A  (data-space XOR masks)
  uint32_t r[5];     // parity masks = rows of A^-1 (a0/a1 orientation)
  int32_t  fp[10];   // free (non-pivot) bit positions for coset enumeration
  int32_t  csbase;   // base index into cos/sin tables (= layer*14 + first wire)
  int32_t  k;        // #wires folded into this pass (5 or 4)
};
struct CircuitDesc {
  PassDesc p[9];             // 3 layers x (5,5,4) wires
  uint32_t zmask[NWIRES];    // final measurement parity masks (rows of A_final^-1)
};

// ---------------- CDNA5 feature probes ----------------
#if defined(__has_builtin)
#  if __has_builtin(__builtin_amdgcn_global_load_async_to_lds_b128)
#    define HAVE_ASYNC_B128 1
#  endif
#  if __has_builtin(__builtin_amdgcn_s_wait_asynccnt)
#    define HAVE_WAIT_ASYNC 1
#  endif
#  if __has_builtin(__builtin_amdgcn_wmma_f32_16x16x4_f32)
#    define HAVE_WMMA_F32X4 1
#  endif
#endif

#define AS_GLOBAL __attribute__((address_space(1)))
#define AS_LOCAL  __attribute__((address_space(3)))
typedef int   v4i __attribute__((ext_vector_type(4)));
typedef float v2f __attribute__((ext_vector_type(2)));
typedef float v8f __attribute__((ext_vector_type(8)));

// ---------------- scalar pass: K wire rotations in registers ----------------
template<int K, bool FINAL>
__device__ __forceinline__
float do_pass(const PassDesc& P,
              const float* __restrict__ csC, const float* __restrict__ csS,
              float* __restrict__ ldsR, float* __restrict__ ldsI,
              const uint32_t* __restrict__ zm, const float* __restrict__ hwL,
              int tid)
{
  constexpr int NS   = 1 << K;            // coset size per thread
  constexpr int NF   = NWIRES - K;        // free bits
  constexpr int REPS = (DIM >> K) / NT;   // cosets per thread

  float acc = 0.0f;

  uint32_t mm[K], rr[K];
  float    cc[K], ss[K];
#pragma unroll
  for (int i = 0; i < K; ++i) {
    mm[i] = P.m[i];
    rr[i] = P.r[i];
    cc[i] = csC[P.csbase + i];
    ss[i] = csS[P.csbase + i];
  }

#pragma unroll
  for (int rep = 0; rep < REPS; ++rep) {
    const uint32_t t = (uint32_t)tid + (uint32_t)rep * NT;
    uint32_t base = 0u;
#pragma unroll
    for (int i = 0; i < NF; ++i) base |= ((t >> i) & 1u) << P.fp[i];
    // orientation normalization: re-pick the representative inside the coset so
    // that all a0/a1 parities are zero (r_i . m_j = delta_ij)
#pragma unroll
    for (int i = 0; i < K; ++i)
      if (__popc(base & rr[i]) & 1) base ^= mm[i];

    uint32_t addr[NS];
    float ar[NS], ai[NS];
#pragma unroll
    for (int u = 0; u < NS; ++u) {
      uint32_t cmb = 0u;
#pragma unroll
      for (int i = 0; i < K; ++i) if (u & (1u << i)) cmb ^= mm[i];
      addr[u] = base ^ cmb;
      ar[u] = ldsR[addr[u]];
      ai[u] = ldsI[addr[u]];
    }

#pragma unroll
    for (int i = 0; i < K; ++i) {
      const float c = cc[i];
      const float s = ss[i];
#pragma unroll
      for (int u = 0; u < NS; ++u) {
        if ((u & (1u << i)) == 0) {
          const int v = u | (1u << i);
          const float x0r = ar[u], x1r = ar[v];
          const float x0i = ai[u], x1i = ai[v];
          ar[u] = c * x0r - s * x1r;
          ar[v] = s * x0r + c * x1r;
          ai[u] = c * x0i - s * x1i;
          ai[v] = s * x0i + c * x1i;
        }
      }
    }

    if constexpr (FINAL) {
#pragma unroll
      for (int u = 0; u < NS; ++u) {
        float coeff = 0.0f;
#pragma unroll
        for (int q = 0; q < NWIRES; ++q) {
          const float wq = hwL[q];
          coeff += (__popc(addr[u] & zm[q]) & 1) ? -wq : wq;
        }
        acc += (ar[u] * ar[u] + ai[u] * ai[u]) * coeff;
      }
    } else {
#pragma unroll
      for (int u = 0; u < NS; ++u) {
        ldsR[addr[u]] = ar[u];
        ldsI[addr[u]] = ai[u];
      }
    }
  }
  return acc;
}

#if defined(HAVE_WMMA_F32X4)
// D[u][v] = prod_i R_i[u_i][v_i],  R = [[c,-s],[s,c]]  (kron of 4 RY(2x2))
__device__ __forceinline__
float dmat_entry(int mrow, int kcol, const float* cc, const float* ss)
{
  float v = 1.0f;
#pragma unroll
  for (int i = 0; i < 4; ++i) {
    const int mb = (mrow >> i) & 1, kb = (kcol >> i) & 1;
    const float f = (mb == kb) ? cc[i] : ((mb == 1) ? ss[i] : -ss[i]);
    v *= f;
  }
  return v;
}

// ---------------- WMMA pass: 4 wire rotations = 16x16 f32 matmul ------------
// ISA 7.12.2 32-bit layouts: A(16x4): lane=M, K=2*(lane>=16)+vgpr;
// B(4x16): lane=N(mod16), same K split; C/D(16x16): lane=N, vgpr g -> M=g+8*(lane>=16)
template<bool FINAL>
__device__ __forceinline__
float do_pass4_wmma(const PassDesc& P,
                    const float* __restrict__ csC, const float* __restrict__ csS,
                    float* __restrict__ ldsR, float* __restrict__ ldsI,
                    const uint32_t* __restrict__ zm, const float* __restrict__ hwL,
                    int tid)
{
  uint32_t mm[4], rr[4];
  float    cc[4], ss[4];
#pragma unroll
  for (int i = 0; i < 4; ++i) {
    mm[i] = P.m[i];
    rr[i] = P.r[i];
    cc[i] = csC[P.csbase + i];
    ss[i] = csS[P.csbase + i];
  }

  const int lane = tid & 31;
  const int n    = lane & 15;           // this lane's column (coset)
  const int hh   = (lane >> 4) << 1;    // K sub-offset: 0 or 2
  const int hi8  = (lane >> 4) << 3;    // C/D row offset: 0 or 8
  const int wave = tid >> 5;

  // A fragments: shared by every batch and by real/imag (rotation is real)
  v2f aFrag[4];
#pragma unroll
  for (int j = 0; j < 4; ++j) {
    aFrag[j].x = dmat_entry(n, 4 * j + hh + 0, cc, ss);
    aFrag[j].y = dmat_entry(n, 4 * j + hh + 1, cc, ss);
  }

  constexpr int NCOSET   = DIM >> 4;               // 1024
  constexpr int PER_WAVE = NCOSET / (NT / 32);     // 64
  constexpr int NBATCH   = PER_WAVE / 16;          // 4

  float acc = 0.0f;
#pragma unroll
  for (int bt = 0; bt < NBATCH; ++bt) {
    const uint32_t t = (uint32_t)(wave * PER_WAVE + bt * 16 + n);
    uint32_t base = 0u;
#pragma unroll
    for (int i = 0; i < 10; ++i) base |= ((t >> i) & 1u) << P.fp[i];
    // orientation normalization -> one shared D matrix for all 16 columns
#pragma unroll
    for (int i = 0; i < 4; ++i)
      if (__popc(base & rr[i]) & 1) base ^= mm[i];

    v8f cr = {}; v8f ci = {};
#pragma unroll
    for (int j = 0; j < 4; ++j) {
      const int k0 = 4 * j + hh, k1 = k0 + 1;
      uint32_t c0 = 0u, c1 = 0u;
#pragma unroll
      for (int i = 0; i < 4; ++i) {
        if ((k0 >> i) & 1) c0 ^= mm[i];
        if ((k1 >> i) & 1) c1 ^= mm[i];
      }
      const uint32_t a0 = base ^ c0, a1 = base ^ c1;
      v2f br; br.x = ldsR[a0]; br.y = ldsR[a1];
      v2f bi; bi.x = ldsI[a0]; bi.y = ldsI[a1];
      cr = __builtin_amdgcn_wmma_f32_16x16x4_f32(false, aFrag[j], false, br,
                                                 (short)0, cr, false, false);
      ci = __builtin_amdgcn_wmma_f32_16x16x4_f32(false, aFrag[j], false, bi,
                                                 (short)0, ci, false, false);
    }

#pragma unroll
    for (int g = 0; g < 8; ++g) {
      const int mrow = g + hi8;
      uint32_t cm = 0u;
#pragma unroll
      for (int i = 0; i < 4; ++i) if ((mrow >> i) & 1) cm ^= mm[i];
      const uint32_t ad = base ^ cm;
      if constexpr (FINAL) {
        float coeff = 0.0f;
#pragma unroll
        for (int q = 0; q < NWIRES; ++q) {
          const float wq = hwL[q];
          coeff += (__popc(ad & zm[q]) & 1) ? -wq : wq;
        }
        acc += (cr[g] * cr[g] + ci[g] * ci[g]) * coeff;
      } else {
        ldsR[ad] = cr[g];
        ldsI[ad] = ci[g];
      }
    }
  }
  return acc;
}
#endif // HAVE_WMMA_F32X4

// ---------------- main kernel: one workgroup per batch state ----------------
__launch_bounds__(NT)
__global__ void qsim_kernel(const float* __restrict__ sr,
                            const float* __restrict__ si,
                            const float* __restrict__ w,
                            const float* __restrict__ hw,
                            const float* __restrict__ hb,
                            float* __restrict__ out,
                            CircuitDesc cd)
{
  extern __shared__ float smem[];
  float* ldsR = smem;                       // DIM floats (64 KB)
  float* ldsI = smem + DIM;                 // DIM floats (64 KB)
  float* csC  = smem + 2 * DIM;             // 42
  float* csS  = csC + NLAYERS * NWIRES;     // 42
  float* hwL  = csS + NLAYERS * NWIRES;     // 14
  float* red  = hwL + NWIRES;               // 16 (per-wave partials)

  const int tid = threadIdx.x;
  const int b   = blockIdx.x;

  // ---- stage state into LDS (async DMA path on CDNA5) ----
  const float4* gr = reinterpret_cast<const float4*>(sr + (size_t)b * DIM);
  const float4* gi = reinterpret_cast<const float4*>(si + (size_t)b * DIM);
#if defined(HAVE_ASYNC_B128)
#pragma unroll
  for (int c = 0; c < (DIM / 4) / NT; ++c) {
    const int f = tid + c * NT;
    __builtin_amdgcn_global_load_async_to_lds_b128(
        (AS_GLOBAL v4i*)(gr + f), (AS_LOCAL v4i*)(ldsR + 4 * f), 0, 0);
    __builtin_amdgcn_global_load_async_to_lds_b128(
        (AS_GLOBAL v4i*)(gi + f), (AS_LOCAL v4i*)(ldsI + 4 * f), 0, 0);
  }
#else
#pragma unroll
  for (int c = 0; c < (DIM / 4) / NT; ++c) {
    const int f = tid + c * NT;
    reinterpret_cast<float4*>(ldsR)[f] = gr[f];
    reinterpret_cast<float4*>(ldsI)[f] = gi[f];
  }
#endif

  // ---- overlap: fold PPL weights, cos/sin tables, head weights ----
  if (tid < NLAYERS * NWIRES) {
    const int l = tid / NWIRES, q = tid % NWIRES;
    float ang = 0.0f;
#pragma unroll
    for (int p = 0; p < PPL; ++p) ang += w[(l * PPL + p) * NWIRES + q];
    const float h = 0.5f * ang;
    csC[tid] = cosf(h);
    csS[tid] = sinf(h);
  } else if (tid >= 64 && tid < 64 + NWIRES) {
    hwL[tid - 64] = hw[tid - 64];
  }

#if defined(HAVE_ASYNC_B128)
#  if defined(HAVE_WAIT_ASYNC)
  __builtin_amdgcn_s_wait_asynccnt(0);
#  else
  asm volatile("s_wait_asynccnt 0" ::: "memory");
#  endif
#endif
  __syncthreads();

  // ---- 3 layers x (5+5+4) wires; CNOTs are free (folded into masks) ----
  float acc;
  for (int l = 0; l < NLAYERS; ++l) {
    do_pass<5, false>(cd.p[3 * l + 0], csC, csS, ldsR, ldsI, nullptr, nullptr, tid);
    __syncthreads();
    do_pass<5, false>(cd.p[3 * l + 1], csC, csS, ldsR, ldsI, nullptr, nullptr, tid);
    __syncthreads();
    if (l < NLAYERS - 1) {
#if defined(HAVE_WMMA_F32X4)
      do_pass4_wmma<false>(cd.p[3 * l + 2], csC, csS, ldsR, ldsI, nullptr, nullptr, tid);
#else
      do_pass<4, false>(cd.p[3 * l + 2], csC, csS, ldsR, ldsI, nullptr, nullptr, tid);
#endif
      __syncthreads();
    }
  }
  // final pass fuses measurement + head dot product
#if defined(HAVE_WMMA_F32X4)
  acc = do_pass4_wmma<true>(cd.p[8], csC, csS, ldsR, ldsI, cd.zmask, hwL, tid);
#else
  acc = do_pass<4, true>(cd.p[8], csC, csS, ldsR, ldsI, cd.zmask, hwL, tid);
#endif

  // ---- block reduction ----
#pragma unroll
  for (int o = 16; o > 0; o >>= 1) acc += __shfl_xor(acc, o, 32);
  if ((tid & 31) == 0) red[tid >> 5] = acc;
  __syncthreads();
  if (tid == 0) {
    float tot = 0.0f;
#pragma unroll
    for (int i = 0; i < NT / 32; ++i) tot += red[i];
    out[b] = tot + hb[0];
  }
}

// ---------------- host: fold the CNOT network into GF(2) masks ----------------
static void build_circuit(CircuitDesc& cd)
{
  uint32_t cols[NWIRES], rows[NWIRES];
  for (int bb = 0; bb < NWIRES; ++bb) { cols[bb] = 1u << bb; rows[bb] = 1u << bb; }

  uint32_t M[NLAYERS][NWIRES], R[NLAYERS][NWIRES];
  for (int l = 0; l < NLAYERS; ++l) {
    for (int q = 0; q < NWIRES; ++q) {       // snapshot before this layer's CNOTs
      const int bb = NWIRES - 1 - q;         // wire q <-> bit 13-q (wire0 = MSB)
      M[l][q] = cols[bb];
      R[l][q] = rows[bb];
    }
    for (int j = 0; j < NWIRES; ++j) {       // CNOT(j, (j+1)%14): bit_t ^= bit_c
      const int cb = NWIRES - 1 - j;
      const int tb = NWIRES - 1 - ((j + 1) % NWIRES);
      cols[cb] ^= cols[tb];                  // A <- A*Mcnot
      rows[tb] ^= rows[cb];                  // A^-1 <- Mcnot*A^-1
    }
  }
  for (int q = 0; q < NWIRES; ++q) cd.zmask[q] = rows[NWIRES - 1 - q];

  const int q0s[3] = {0, 5, 10};
  const int ks[3]  = {5, 5, 4};
  int pi = 0;
  for (int l = 0; l < NLAYERS; ++l) {
    for (int pp = 0; pp < 3; ++pp) {
      PassDesc& P = cd.p[pi++];
      const int k = ks[pp], q0 = q0s[pp];
      for (int i = 0; i < 5; ++i) {
        P.m[i] = (i < k) ? M[l][q0 + i] : 0u;
        P.r[i] = (i < k) ? R[l][q0 + i] : 0u;
      }
      P.csbase = l * NWIRES + q0;
      P.k = k;
      uint32_t g[5]; int pb[5];
      for (int i = 0; i < k; ++i) g[i] = P.m[i];
      uint32_t pivmask = 0;
      for (int i = 0; i < k; ++i) {
        for (int it = 0; it < k; ++it)
          for (int tprev = 0; tprev < i; ++tprev)
            if (g[i] & (1u << pb[tprev])) g[i] ^= g[tprev];
        if (!g[i]) { pb[i] = 0; continue; }
        const int h = 31 - __builtin_clz(g[i]);
        pb[i] = h;
        pivmask |= (1u << h);
      }
      int nf = 0;
      for (int bb = 0; bb < NWIRES; ++bb)
        if (!(pivmask & (1u << bb))) P.fp[nf++] = bb;
      for (; nf < 10; ++nf) P.fp[nf] = 0;
    }
  }
}

extern "C" void kernel_launch(void* const* d_in, const int* in_sizes, int n_in,
                              void* d_out, int out_size, void* d_ws, size_t ws_size,
                              hipStream_t stream)
{
  (void)in_sizes; (void)n_in; (void)d_ws; (void)ws_size; (void)out_size;
  const float* sr = (const float*)d_in[0];   // (1024, 16384)
  const float* si = (const float*)d_in[1];   // (1024, 16384)
  const float* w  = (const float*)d_in[2];   // (3, 3, 14)
  const float* hw = (const float*)d_in[3];   // (1, 14)
  const float* hb = (const float*)d_in[4];   // (1,)
  float* out = (float*)d_out;                // (1024,)

  CircuitDesc cd;
  build_circuit(cd);                         // deterministic, weight-independent

  const size_t smem_bytes =
      (size_t)(2 * DIM + 2 * NLAYERS * NWIRES + NWIRES + NT / 32) * sizeof(float);

  qsim_kernel<<<dim3(BATCH), dim3(NT), smem_bytes, stream>>>(
      sr, si, w, hw, hb, out, cd);
}